// snn_mnist_53377853554886
// MI455X (gfx1250) — compile-verified
//
#include <hip/hip_runtime.h>

// ---- problem constants (match reference) ----
static constexpr int T_STEPS = 200;
static constexpr int B_SZ    = 256;
static constexpr int IN_SZ   = 784;
static constexpr int N_SZ    = 400;

typedef __attribute__((ext_vector_type(2))) float v2f;
typedef __attribute__((ext_vector_type(8))) float v8f;

// Full-precision f32 WMMA: D[16x16] += A[16x4] * B[4x16]
// A: lane L holds row  M = L%16, K = 2*(L/16)+{0,1}  (contiguous K pair)
// B: lane L holds col  N = L%16, K = 2*(L/16)+{0,1}  (contiguous K pair)
// C/D: lane L, vgpr v -> (M = v + 8*(L/16), N = L%16)
static __device__ __forceinline__ v8f wmma_f32_k4(v2f a, v2f b, v8f c) {
  return __builtin_amdgcn_wmma_f32_16x16x4_f32(
      /*neg_a=*/false, a, /*neg_b=*/false, b,
      /*c_mod=*/(short)0, c, /*reuse_a=*/false, /*reuse_b=*/false);
}

// ---------------------------------------------------------------------------
// Kernel A: cur = img_t @ W^T  (WMMA), then LIF update + records + argmin-spike
// grid: 50 blocks x 256 thr = 400 waves = (B/16=16) x (N/16=25) tiles
// ---------------------------------------------------------------------------
__global__ void k_forward(const float* __restrict__ img_t,   // [B,IN]
                          const float* __restrict__ W,       // [N,IN]
                          float* __restrict__ syn,           // [B,N] state
                          float* __restrict__ mem,           // [B,N] state
                          float* __restrict__ mem_rec_t,     // [B,N] out
                          float* __restrict__ spk_rec_t,     // [B,N] out
                          int* __restrict__ win_cur,         // [B] this step
                          int* __restrict__ flag_cur,        // [1] this step
                          int* __restrict__ win_next,        // [B] reset for t+1
                          int* __restrict__ flag_next) {     // [1] reset for t+1
  // reset the *other* ping-pong slot (untouched this step -> race-free)
  if (blockIdx.x == 0) {
    if ((int)threadIdx.x < B_SZ) win_next[threadIdx.x] = 0x7FFFFFFF;
    if (threadIdx.x == 0) *flag_next = 0;
  }

  const int wave = threadIdx.x >> 5;
  const int lane = threadIdx.x & 31;
  const int m    = lane & 15;
  const int half = lane >> 4;
  const int tile = blockIdx.x * 8 + wave;     // 0..399
  const int bt   = tile / 25;                 // batch tile 0..15
  const int nt   = tile % 25;                 // neuron tile 0..24
  const int b0   = bt * 16, n0 = nt * 16;

  const float* aRow = img_t + (size_t)(b0 + m) * IN_SZ + 2 * half;
  const float* bRow = W     + (size_t)(n0 + m) * IN_SZ + 2 * half;

  v8f acc = {};
  for (int kc = 0; kc < IN_SZ; kc += 16) {          // 49 chunks of K=16
    __builtin_prefetch(aRow + kc + 64, 0, 3);       // near-cache prefetch
    __builtin_prefetch(bRow + kc + 64, 0, 3);
#pragma unroll
    for (int ku = 0; ku < 16; ku += 4) {
      v2f a = *(const v2f*)(aRow + kc + ku);
      v2f b = *(const v2f*)(bRow + kc + ku);
      acc = wmma_f32_k4(a, b, acc);
    }
  }

  bool any_spk = false;
  const int ni = n0 + m;
#pragma unroll
  for (int v = 0; v < 8; ++v) {
    const int bi = b0 + v + 8 * half;
    const size_t idx = (size_t)bi * N_SZ + ni;
    const float cur = acc[v];
    float s  = syn[idx];
    float mv = mem[idx];
    const float reset = (mv > 1.0f) ? 1.0f : 0.0f;   // from incoming mem
    s  = 0.9f * s + cur;                             // ALPHA
    mv = 0.8f * mv + s - reset;                      // BETA, THRESH=1
    const float spk = (mv > 1.0f) ? 1.0f : 0.0f;
    syn[idx]       = s;
    mem[idx]       = mv;
    mem_rec_t[idx] = mv;
    spk_rec_t[idx] = spk;
    if (spk > 0.0f) { atomicMin(&win_cur[bi], ni); any_spk = true; }
  }
  if (any_spk) atomicOr(flag_cur, 1);
}

// ---------------------------------------------------------------------------
// Kernel B (fused): lateral inhibition + post-trace + pre-trace double-buffer
// grid: 784 x 256 = B*IN threads exactly (first B*N also do syn/post work)
// ---------------------------------------------------------------------------
__global__ void k_inhibit_post_trace(float* __restrict__ syn,
                                     float* __restrict__ post,
                                     const float* __restrict__ spk_rec_t,
                                     const int* __restrict__ win_cur,
                                     const int* __restrict__ flag_cur,
                                     const float* __restrict__ tr_src,  // tr_t
                                     float* __restrict__ tr_dst,       // tr_{t+1}
                                     const float* __restrict__ img_t) {
  const int idx = blockIdx.x * blockDim.x + threadIdx.x;   // < B*IN exactly
  if (idx < B_SZ * N_SZ) {
    const int bi = idx / N_SZ;
    const int ni = idx - bi * N_SZ;
    if (*flag_cur) {
      int w = win_cur[bi];
      if (w == 0x7FFFFFFF) w = 0;     // argmax of all-zero row is 0
      if (ni != w) syn[idx] -= 0.1f;  // LI_STRENGTH
    }
    post[idx] = 0.9f * post[idx] + spk_rec_t[idx];         // BETA_MINUS
  }
  // STDP (kernel C) reads tr_src; we produce tr_{t+1} into the other slot
  tr_dst[idx] = 0.9f * tr_src[idx] + img_t[idx];           // BETA_PLUS
}

// ---------------------------------------------------------------------------
// Kernel C: STDP  W = clip(W + A+ * spk^T@tr - A- * post^T@img, 0, 1)
// grid: 175 blocks x 224 thr (7 waves). Block = one nt (16 neurons) x 7 it.
// spk/post column panels [256 x 16] staged once per block into LDS via the
// CDNA5 async global->LDS DMA path (ASYNCcnt).
// ---------------------------------------------------------------------------
__global__ void __launch_bounds__(224)
k_stdp(const float* __restrict__ spk_t,    // [B,N]
       const float* __restrict__ post,     // [B,N] (updated)
       const float* __restrict__ tr,       // [B,IN] (tr_t, pre-update)
       const float* __restrict__ img_t,    // [B,IN]
       float* __restrict__ W) {            // [N,IN]
  __shared__ float sSpk [B_SZ * 16];       // [k][c] : 16 KB
  __shared__ float sPost[B_SZ * 16];       // [k][c] : 16 KB

  const int wave = threadIdx.x >> 5;               // 0..6
  const int lane = threadIdx.x & 31;
  const int m    = lane & 15;
  const int half = lane >> 4;
  const int nt   = blockIdx.x / 7;                 // 0..24
  const int it   = (blockIdx.x % 7) * 7 + wave;    // 0..48
  const int n0   = nt * 16, i0 = it * 16;

  // ---- async stage spk/post panels (columns n0..n0+15, all 256 batch rows)
  for (int idx = threadIdx.x; idx < B_SZ * 16; idx += 224) {
    const int k = idx >> 4;
    const int c = idx & 15;
    const unsigned long long gs =
        (unsigned long long)(uintptr_t)(spk_t + (size_t)k * N_SZ + n0 + c);
    const unsigned long long gp =
        (unsigned long long)(uintptr_t)(post  + (size_t)k * N_SZ + n0 + c);
    const unsigned ls = (unsigned)(uintptr_t)(&sSpk[idx]);   // LDS byte offset
    const unsigned lp = (unsigned)(uintptr_t)(&sPost[idx]);
    asm volatile("global_load_async_to_lds_b32 %0, %1, off"
                 :: "v"(ls), "v"(gs) : "memory");
    asm volatile("global_load_async_to_lds_b32 %0, %1, off"
                 :: "v"(lp), "v"(gp) : "memory");
  }
  asm volatile("s_wait_asynccnt 0" ::: "memory");  // drain this wave's DMAs
  __syncthreads();                                 // panels visible block-wide

  // ---- WMMA accumulation over K = batch (B operands straight from global)
  v8f accP = {};   // spk^T @ tr
  v8f accM = {};   // post^T @ img
  const int j = i0 + m;                            // B-matrix column
  for (int kb = 0; kb < B_SZ; kb += 4) {
    const int k0 = kb + 2 * half;
    v2f aP, aM, bP, bM;
    aP.x = sSpk [k0 * 16 + m];
    aP.y = sSpk [(k0 + 1) * 16 + m];
    aM.x = sPost[k0 * 16 + m];
    aM.y = sPost[(k0 + 1) * 16 + m];
    bP.x = tr   [(size_t)k0 * IN_SZ + j];
    bP.y = tr   [(size_t)(k0 + 1) * IN_SZ + j];
    bM.x = img_t[(size_t)k0 * IN_SZ + j];
    bM.y = img_t[(size_t)(k0 + 1) * IN_SZ + j];
    accP = wmma_f32_k4(aP, bP, accP);
    accM = wmma_f32_k4(aM, bM, accM);
  }
#pragma unroll
  for (int v = 0; v < 8; ++v) {
    const int nr = n0 + v + 8 * half;
    const size_t idx = (size_t)nr * IN_SZ + j;
    float w = W[idx] + 1.0e-3f * accP[v] - 1.0e-3f * accM[v];
    w = fminf(fmaxf(w, 0.0f), 1.0f);
    W[idx] = w;
  }
}

// ---------------------------------------------------------------------------
// Init: zero all state, set argmin sentinels (slot 0; slot 1 reset by k_forward)
// ---------------------------------------------------------------------------
__global__ void k_init(float* __restrict__ syn, float* __restrict__ mem,
                       float* __restrict__ post, float* __restrict__ tr0,
                       float* __restrict__ tr1,
                       int* __restrict__ win, int* __restrict__ flags) {
  const int idx = blockIdx.x * blockDim.x + threadIdx.x;   // 200704 threads
  if (idx < B_SZ * N_SZ) { syn[idx] = 0.0f; mem[idx] = 0.0f; post[idx] = 0.0f; }
  if (idx < B_SZ * IN_SZ) { tr0[idx] = 0.0f; tr1[idx] = 0.0f; }
  if (idx < 2 * B_SZ) win[idx] = 0x7FFFFFFF;
  if (idx < 2) flags[idx] = 0;
}

extern "C" void kernel_launch(void* const* d_in, const int* in_sizes, int n_in,
                              void* d_out, int out_size, void* d_ws, size_t ws_size,
                              hipStream_t stream) {
  (void)in_sizes; (void)n_in; (void)out_size; (void)ws_size;
  const float* image = (const float*)d_in[0];  // [T,B,IN]
  const float* W0    = (const float*)d_in[1];  // [N,IN]

  float* out     = (float*)d_out;
  float* mem_rec = out;                                       // [T,B,N]
  float* spk_rec = out + (size_t)T_STEPS * B_SZ * N_SZ;       // [T,B,N]
  float* W       = out + 2 * (size_t)T_STEPS * B_SZ * N_SZ;   // [N,IN] in place

  // workspace: syn | mem | post | tr0 | tr1 | win[2*B] | flags[2]
  float* syn  = (float*)d_ws;
  float* mem  = syn  + (size_t)B_SZ * N_SZ;
  float* post = mem  + (size_t)B_SZ * N_SZ;
  float* trb[2];
  trb[0] = post + (size_t)B_SZ * N_SZ;
  trb[1] = trb[0] + (size_t)B_SZ * IN_SZ;
  int* win   = (int*)(trb[1] + (size_t)B_SZ * IN_SZ);
  int* flags = win + 2 * B_SZ;

  hipMemcpyAsync(W, W0, (size_t)N_SZ * IN_SZ * sizeof(float),
                 hipMemcpyDeviceToDevice, stream);
  k_init<<<784, 256, 0, stream>>>(syn, mem, post, trb[0], trb[1], win, flags);

  for (int t = 0; t < T_STEPS; ++t) {
    const int s = t & 1;                       // tr_t lives in slot s
    const float* img_t = image + (size_t)t * B_SZ * IN_SZ;
    float* mem_t = mem_rec + (size_t)t * B_SZ * N_SZ;
    float* spk_t = spk_rec + (size_t)t * B_SZ * N_SZ;

    k_forward<<<50, 256, 0, stream>>>(img_t, W, syn, mem, mem_t, spk_t,
                                      win + s * B_SZ, flags + s,
                                      win + (1 - s) * B_SZ, flags + (1 - s));
    k_inhibit_post_trace<<<784, 256, 0, stream>>>(
        syn, post, spk_t, win + s * B_SZ, flags + s,
        trb[s], trb[1 - s], img_t);
    k_stdp<<<175, 224, 0, stream>>>(spk_t, post, trb[s], img_t, W);
  }
}